// MeanShift_9191230013532
// MI455X (gfx1250) — compile-verified
//
#include <hip/hip_runtime.h>
#include <stdint.h>

// Mean-shift + NMS + softmax assignment for MI455X (gfx1250, wave32).
// All N x N x 32 matmuls run on V_WMMA_F32_16X16X4_F32 (fp32 in, fp32 acc),
// fused so no N^2 intermediate is ever materialized except the mandatory
// 256MB assignment output (one streaming HBM write).

#define NPTS 8192
#define DIM  32
#define DAMPING 0.5f
#define TOL2 1e-8f            // (1e-4)^2, compare on squared shift

typedef float v2f __attribute__((ext_vector_type(2)));
typedef float v8f __attribute__((ext_vector_type(8)));

__device__ __forceinline__ v8f wmma_f32(v2f a, v2f b, v8f c) {
  // D = A(16x4,f32) * B(4x16,f32) + C(16x16,f32)
  return __builtin_amdgcn_wmma_f32_16x16x4_f32(false, a, false, b, (short)0, c,
                                               false, false);
}

// Load a 16x32 row tile of `A` (row-major, stride DIM) into WMMA A-layout
// chunks a[0..7] (K=4 each) and produce xxv[v] = ||row (v+8*half)||^2.
__device__ __forceinline__ void load_a16(const float* __restrict__ A, int row0,
                                         int lane, v2f a[8], float xxv[8]) {
  const int m = lane & 15, h = lane >> 4;
  const float* ap = A + (size_t)(row0 + m) * DIM + 2 * h;
  float p = 0.f;
#pragma unroll
  for (int c = 0; c < 8; ++c) {
    a[c] = *(const v2f*)(ap + 4 * c);
    p += a[c].x * a[c].x + a[c].y * a[c].y;
  }
  p += __shfl_xor(p, 16, 32);  // full ||row (lane&15)||^2
#pragma unroll
  for (int v = 0; v < 8; ++v) xxv[v] = __shfl(p, v + 8 * h, 32);
}

// Squared distances between the preloaded 16-row A tile and rows
// [j0, j0+16) of B (row-major, stride DIM). Returns C-layout v8f:
// element v <-> (row = row0 + v + 8*half, col = j0 + (lane&15)).
__device__ __forceinline__ v8f sq_tile16(const float* __restrict__ B,
                                         const v2f a[8], const float xxv[8],
                                         int j0, int lane) {
  const int n = lane & 15, h = lane >> 4;
  const float* bp = B + (size_t)(j0 + n) * DIM + 2 * h;
  v2f b[8];
  float cc = 0.f;
#pragma unroll
  for (int c = 0; c < 8; ++c) {
    b[c] = *(const v2f*)(bp + 4 * c);
    cc += b[c].x * b[c].x + b[c].y * b[c].y;
  }
  cc += __shfl_xor(cc, 16, 32);  // ||col row||^2
  v8f dot = {};
#pragma unroll
  for (int c = 0; c < 8; ++c) dot = wmma_f32(a[c], b[c], dot);
  v8f sq;
#pragma unroll
  for (int v = 0; v < 8; ++v) {
    float s = xxv[v] + cc - 2.f * dot[v];
    sq[v] = s > 0.f ? s : 0.f;
  }
  return sq;
}

// One mean-shift iteration: NC[i] = sum_j exp(-g*d2(x_i,c_j)) * x_j / sum_j w.
// Fused: never materializes w. 8 waves/block, one 16-row tile per wave.
__global__ __launch_bounds__(256) void k_iter(const float* __restrict__ X,
                                              const float* __restrict__ C,
                                              float* __restrict__ NC,
                                              const float* __restrict__ sigp) {
  __shared__ float wb[8][16 * 20];  // per-wave C-layout -> A-layout staging
  const int lane = threadIdx.x & 31, wave = threadIdx.x >> 5;
  const int row0 = (blockIdx.x * 8 + wave) * 16;
  const int n = lane & 15, h = lane >> 4;
  const float sg = sigp[0];
  const float gamma = 1.f / (2.f * sg * sg);

  v2f a[8];
  float xxv[8];
  load_a16(X, row0, lane, a, xxv);

  v8f acc0 = {}, acc1 = {}, accw = {};
  float* wbw = wb[wave];

  for (int j0 = 0; j0 < NPTS; j0 += 16) {
    v8f sq = sq_tile16(C, a, xxv, j0, lane);
    v8f wv;
#pragma unroll
    for (int v = 0; v < 8; ++v) wv[v] = __expf(-gamma * sq[v]);
    accw += wv;
    // transpose w tile to A-layout through LDS (same-wave DS is in-order)
#pragma unroll
    for (int v = 0; v < 8; ++v) wbw[(v + 8 * h) * 20 + n] = wv[v];
    v2f aw[4];
#pragma unroll
    for (int c = 0; c < 4; ++c)
      aw[c] = *(const v2f*)&wbw[n * 20 + 4 * c + 2 * h];
    // numerator += w(16x16) @ x_j(16x32), split into two N=16 halves
#pragma unroll
    for (int c = 0; c < 4; ++c) {
      const float* xr0 = X + (size_t)(j0 + 4 * c + 2 * h) * DIM;
      const float* xr1 = xr0 + DIM;
      v2f b0 = {xr0[n], xr1[n]};
      v2f b1 = {xr0[16 + n], xr1[16 + n]};
      acc0 = wmma_f32(aw[c], b0, acc0);
      acc1 = wmma_f32(aw[c], b1, acc1);
    }
  }
  // denominator: reduce accw over the 16 column-lanes of each half
#pragma unroll
  for (int v = 0; v < 8; ++v) {
    float d = accw[v];
    d += __shfl_xor(d, 1, 32);
    d += __shfl_xor(d, 2, 32);
    d += __shfl_xor(d, 4, 32);
    d += __shfl_xor(d, 8, 32);
    float r = 1.f / d;
    NC[(size_t)(row0 + v + 8 * h) * DIM + n] = acc0[v] * r;
    NC[(size_t)(row0 + v + 8 * h) * DIM + 16 + n] = acc1[v] * r;
  }
}

__global__ void k_init(const float* __restrict__ X, float* __restrict__ C,
                       int* __restrict__ scores, int* __restrict__ flag) {
  int i = blockIdx.x * blockDim.x + threadIdx.x;
  if (i < NPTS * DIM) C[i] = X[i];
  if (i < NPTS) scores[i] = 0;
  if (i == 0) *flag = 0;
}

__global__ void k_shift(const float* __restrict__ C, const float* __restrict__ NC,
                        float* __restrict__ s2) {
  int i = blockIdx.x * blockDim.x + threadIdx.x;
  if (i >= NPTS) return;
  const float* c = C + (size_t)i * DIM;
  const float* nc = NC + (size_t)i * DIM;
  float s = 0.f;
#pragma unroll
  for (int d = 0; d < DIM; ++d) {
    float df = nc[d] - c[d];
    s += df * df;
  }
  s2[i] = s;
}

__global__ __launch_bounds__(256) void k_flag(const float* __restrict__ s2,
                                              int* __restrict__ flag) {
  __shared__ float red[256];
  float mx = 0.f;
  for (int i = threadIdx.x; i < NPTS; i += 256) mx = fmaxf(mx, s2[i]);
  red[threadIdx.x] = mx;
  __syncthreads();
  for (int s = 128; s > 0; s >>= 1) {
    if (threadIdx.x < (unsigned)s)
      red[threadIdx.x] = fmaxf(red[threadIdx.x], red[threadIdx.x + s]);
    __syncthreads();
  }
  if (threadIdx.x == 0 && red[0] < TOL2) *flag = 1;  // sticky
}

__global__ void k_update(float* __restrict__ C, const float* __restrict__ NC,
                         const int* __restrict__ flag) {
  int i = blockIdx.x * blockDim.x + threadIdx.x;
  if (i < NPTS * DIM && *flag == 0)
    C[i] = DAMPING * C[i] + (1.f - DAMPING) * NC[i];
}

__global__ void k_copy(const float* __restrict__ C, float* __restrict__ o) {
  int i = blockIdx.x * blockDim.x + threadIdx.x;
  if (i < NPTS * DIM) o[i] = C[i];
}

// argmin_j d2(x_i, c_j) -> scores[argmin]++ (first-minimum tie break).
__global__ __launch_bounds__(256) void k_closest(const float* __restrict__ X,
                                                 const float* __restrict__ C,
                                                 int* __restrict__ scores) {
  const int lane = threadIdx.x & 31, wave = threadIdx.x >> 5;
  const int row0 = (blockIdx.x * 8 + wave) * 16;
  const int n = lane & 15;
  v2f a[8];
  float xxv[8];
  load_a16(X, row0, lane, a, xxv);
  float best[8];
  int bidx[8];
#pragma unroll
  for (int v = 0; v < 8; ++v) {
    best[v] = 3.0e38f;
    bidx[v] = 0;
  }
  for (int j0 = 0; j0 < NPTS; j0 += 16) {
    v8f sq = sq_tile16(C, a, xxv, j0, lane);
#pragma unroll
    for (int v = 0; v < 8; ++v) {
      if (sq[v] < best[v]) {
        best[v] = sq[v];
        bidx[v] = j0 + n;
      }
    }
  }
#pragma unroll
  for (int v = 0; v < 8; ++v) {
#pragma unroll
    for (int msk = 1; msk <= 8; msk <<= 1) {
      float ovv = __shfl_xor(best[v], msk, 32);
      int oi = __shfl_xor(bidx[v], msk, 32);
      if (ovv < best[v] || (ovv == best[v] && oi < bidx[v])) {
        best[v] = ovv;
        bidx[v] = oi;
      }
    }
  }
  if (n == 0) {
#pragma unroll
    for (int v = 0; v < 8; ++v) atomicAdd(&scores[bidx[v]], 1);
  }
}

// overlap bitmask: bit(col) of row r set iff d2(c_r,c_col) < sigma^2 && r!=col.
__global__ __launch_bounds__(256) void k_overlap(const float* __restrict__ C,
                                                 uint32_t* __restrict__ ov,
                                                 const float* __restrict__ sigp) {
  const int lane = threadIdx.x & 31, wave = threadIdx.x >> 5;
  const int row0 = (blockIdx.x * 8 + wave) * 16;
  const int n = lane & 15, h = lane >> 4;
  const float thr2 = sigp[0] * sigp[0];
  v2f a[8];
  float xxv[8];
  load_a16(C, row0, lane, a, xxv);
  for (int j0 = 0; j0 < NPTS; j0 += 32) {
    v8f s0 = sq_tile16(C, a, xxv, j0, lane);
    v8f s1 = sq_tile16(C, a, xxv, j0 + 16, lane);
    uint32_t b0[8], b1[8];
#pragma unroll
    for (int v = 0; v < 8; ++v) {
      int r = row0 + v + 8 * h;
      b0[v] = (uint32_t)__ballot((s0[v] < thr2) && (r != (j0 + n)));
      b1[v] = (uint32_t)__ballot((s1[v] < thr2) && (r != (j0 + 16 + n)));
    }
    // lanes 0..15 each assemble + store one 32-col word for local row = lane
    uint32_t m0 = 0, m1 = 0;
#pragma unroll
    for (int v = 0; v < 8; ++v)
      if ((n & 7) == v) {
        m0 = b0[v];
        m1 = b1[v];
      }
    uint32_t sh = (uint32_t)(n >> 3) * 16u;
    uint32_t word = ((m0 >> sh) & 0xFFFFu) | (((m1 >> sh) & 0xFFFFu) << 16);
    if (lane < 16) ov[(size_t)(row0 + n) * (NPTS / 32) + (j0 >> 5)] = word;
  }
}

// Stable rank for argsort(-scores): pos = #{j: s_j>s_i} + #{j<i: s_j==s_i}.
__global__ void k_rank(const int* __restrict__ scores, int* __restrict__ order) {
  int i = blockIdx.x * blockDim.x + threadIdx.x;
  if (i >= NPTS) return;
  int si = scores[i];
  int pos = 0;
  for (int j = 0; j < NPTS; ++j) {
    int sj = scores[j];
    pos += (sj > si) || (sj == si && j < i);
  }
  order[pos] = i;
}

// Sequential greedy NMS (faithful to the scan): single workgroup, LDS bitmask.
__global__ __launch_bounds__(256) void k_nms(const int* __restrict__ order,
                                             const int* __restrict__ scores,
                                             const uint32_t* __restrict__ ov,
                                             uint32_t* __restrict__ keepw) {
  __shared__ uint32_t kw[256];
  const int tid = threadIdx.x;
  kw[tid] = 0xFFFFFFFFu;
  __syncthreads();
  for (int t = 0; t < NPTS; ++t) {
    int idx = order[t];
    bool active = (kw[idx >> 5] >> (idx & 31)) & 1u;  // uniform read
    __syncthreads();
    if (active) {
      kw[tid] &= ~ov[(size_t)idx * (NPTS / 32) + tid];
      if (tid == (idx >> 5) && scores[idx] <= 0) kw[tid] &= ~(1u << (idx & 31));
    }
    __syncthreads();
  }
  keepw[tid] = kw[tid];
}

// Softmax pass 1: per-row running (max, sumexp) over kept columns.
__global__ __launch_bounds__(256) void k_pass1(const float* __restrict__ X,
                                               const float* __restrict__ C,
                                               const uint32_t* __restrict__ keepw,
                                               float* __restrict__ mbuf,
                                               float* __restrict__ lbuf) {
  const int lane = threadIdx.x & 31, wave = threadIdx.x >> 5;
  const int row0 = (blockIdx.x * 8 + wave) * 16;
  const int n = lane & 15, h = lane >> 4;
  v2f a[8];
  float xxv[8];
  load_a16(X, row0, lane, a, xxv);
  const float NEGINF = -__builtin_inff();
  float m[8], l[8];
#pragma unroll
  for (int v = 0; v < 8; ++v) {
    m[v] = NEGINF;
    l[v] = 0.f;
  }
  for (int j0 = 0; j0 < NPTS; j0 += 16) {
    v8f sq = sq_tile16(C, a, xxv, j0, lane);
    uint32_t kwrd = keepw[j0 >> 5];
    bool kept = (kwrd >> ((j0 & 31) + n)) & 1u;
    if (kept) {
#pragma unroll
      for (int v = 0; v < 8; ++v) {
        float cost = -sqrtf(sq[v] + 1e-12f);
        float nm = fmaxf(m[v], cost);
        l[v] = l[v] * __expf(m[v] - nm) + __expf(cost - nm);
        m[v] = nm;
      }
    }
  }
#pragma unroll
  for (int v = 0; v < 8; ++v) {
#pragma unroll
    for (int msk = 1; msk <= 8; msk <<= 1) {
      float om = __shfl_xor(m[v], msk, 32);
      float ol = __shfl_xor(l[v], msk, 32);
      float nm = fmaxf(m[v], om);
      float t0 = (l[v] > 0.f) ? l[v] * __expf(m[v] - nm) : 0.f;
      float t1 = (ol > 0.f) ? ol * __expf(om - nm) : 0.f;
      l[v] = t0 + t1;
      m[v] = (l[v] > 0.f) ? nm : NEGINF;
    }
  }
  if (n == 0) {
#pragma unroll
    for (int v = 0; v < 8; ++v) {
      mbuf[row0 + v + 8 * h] = m[v];
      lbuf[row0 + v + 8 * h] = l[v];
    }
  }
}

// Softmax pass 2: recompute cost, write normalized assignment (256MB stream).
__global__ __launch_bounds__(256) void k_pass2(const float* __restrict__ X,
                                               const float* __restrict__ C,
                                               const uint32_t* __restrict__ keepw,
                                               const float* __restrict__ mbuf,
                                               const float* __restrict__ lbuf,
                                               float* __restrict__ out) {
  const int lane = threadIdx.x & 31, wave = threadIdx.x >> 5;
  const int row0 = (blockIdx.x * 8 + wave) * 16;
  const int n = lane & 15, h = lane >> 4;
  v2f a[8];
  float xxv[8];
  load_a16(X, row0, lane, a, xxv);
  float mr[8], rl[8];
#pragma unroll
  for (int v = 0; v < 8; ++v) {
    int r = row0 + v + 8 * h;
    mr[v] = mbuf[r];
    rl[v] = 1.f / lbuf[r];
  }
  for (int j0 = 0; j0 < NPTS; j0 += 16) {
    v8f sq = sq_tile16(C, a, xxv, j0, lane);
    uint32_t kwrd = keepw[j0 >> 5];
    bool kept = (kwrd >> ((j0 & 31) + n)) & 1u;
#pragma unroll
    for (int v = 0; v < 8; ++v) {
      float val = 0.f;
      if (kept) val = __expf(-sqrtf(sq[v] + 1e-12f) - mr[v]) * rl[v];
      out[(size_t)(row0 + v + 8 * h) * NPTS + j0 + n] = val;
    }
  }
}

__global__ void k_keepout(const uint32_t* __restrict__ keepw,
                          float* __restrict__ o) {
  int i = blockIdx.x * blockDim.x + threadIdx.x;
  if (i < NPTS) o[i] = ((keepw[i >> 5] >> (i & 31)) & 1u) ? 1.f : 0.f;
}

extern "C" void kernel_launch(void* const* d_in, const int* in_sizes, int n_in,
                              void* d_out, int out_size, void* d_ws,
                              size_t ws_size, hipStream_t stream) {
  (void)in_sizes; (void)n_in; (void)out_size; (void)ws_size;
  const float* X = (const float*)d_in[0];
  const float* SIG = (const float*)d_in[1];

  float* out = (float*)d_out;
  float* outC = out;                                   // centroids  (8192*32)
  float* outA = out + (size_t)NPTS * DIM;              // assignment (8192*8192)
  float* outK = outA + (size_t)NPTS * NPTS;            // keep       (8192)

  char* w = (char*)d_ws;                               // ~2.2MB of scratch used
  float* C = (float*)(w);                              // 1MB centroids
  float* NC = (float*)(w + (1u << 20));                // 1MB new centroids
  float* S2 = (float*)(w + (2u << 20));                // 32KB shift^2
  float* MB = (float*)(w + (2u << 20) + 32 * 1024);    // 32KB softmax max
  float* LB = (float*)(w + (2u << 20) + 64 * 1024);    // 32KB softmax sum
  int* SC = (int*)(w + (2u << 20) + 96 * 1024);        // 32KB scores
  int* ORD = (int*)(w + (2u << 20) + 128 * 1024);      // 32KB order
  uint32_t* KW = (uint32_t*)(w + (2u << 20) + 160 * 1024);  // 1KB keep bits
  int* FLAG = (int*)(w + (2u << 20) + 164 * 1024);     // done flag
  // 8MB overlap bitmask lives inside the assignment output region: it is
  // fully written by k_overlap, consumed by k_nms, then overwritten by k_pass2.
  uint32_t* OV = (uint32_t*)outA;

  dim3 b256(256);
  k_init<<<dim3((NPTS * DIM + 255) / 256), b256, 0, stream>>>(X, C, SC, FLAG);
  for (int it = 0; it < 10; ++it) {
    k_iter<<<dim3(NPTS / 128), b256, 0, stream>>>(X, C, NC, SIG);
    k_shift<<<dim3(NPTS / 256), b256, 0, stream>>>(C, NC, S2);
    k_flag<<<dim3(1), b256, 0, stream>>>(S2, FLAG);
    k_update<<<dim3(NPTS * DIM / 256), b256, 0, stream>>>(C, NC, FLAG);
  }
  k_copy<<<dim3(NPTS * DIM / 256), b256, 0, stream>>>(C, outC);
  k_closest<<<dim3(NPTS / 128), b256, 0, stream>>>(X, C, SC);
  k_overlap<<<dim3(NPTS / 128), b256, 0, stream>>>(C, OV, SIG);
  k_rank<<<dim3(NPTS / 256), b256, 0, stream>>>(SC, ORD);
  k_nms<<<dim3(1), b256, 0, stream>>>(ORD, SC, OV, KW);
  k_pass1<<<dim3(NPTS / 128), b256, 0, stream>>>(X, C, KW, MB, LB);
  k_pass2<<<dim3(NPTS / 128), b256, 0, stream>>>(X, C, KW, MB, LB, outA);
  k_keepout<<<dim3(NPTS / 256), b256, 0, stream>>>(KW, outK);
}